// Embed_16020228014144
// MI455X (gfx1250) — compile-verified
//
#include <hip/hip_runtime.h>
#include <hip/hip_bf16.h>

// Embedding row-gather: out[row, :] = W_E[tokens[row], :]
// d_model = 768 f32 = 3072 bytes per row = 192 x 16B.
// One wave32 per row: each lane moves 6 x 16B (lane + 32*i), fully coalesced.
// Loads of W_E stay regular-temporal (192MB L2 nearly holds the 147MB table);
// output stores are non-temporal (global_store_b128 th:NT) so the 96MB of
// streaming writes do not evict the embedding table from L2.

typedef float v4f __attribute__((ext_vector_type(4)));

#define D_MODEL 768
#define F4_PER_ROW (D_MODEL / 4)      // 192
#define F4_PER_LANE (F4_PER_ROW / 32) // 6
#define WAVES_PER_BLOCK 8             // 256 threads

__global__ __launch_bounds__(256) void embed_gather_kernel(
    const int* __restrict__ tokens,
    const float* __restrict__ W_E,
    float* __restrict__ out,
    int n_rows) {
  const int lane = threadIdx.x & 31;
  const int wave = threadIdx.x >> 5;
  const int row  = blockIdx.x * WAVES_PER_BLOCK + wave;
  if (row >= n_rows) return;

  // Wave-uniform token index (single broadcast load per wave).
  const int tok = tokens[row];

  const v4f* __restrict__ src = (const v4f*)(W_E + (size_t)tok * D_MODEL);
  v4f* __restrict__ dst       = (v4f*)(out + (size_t)row * D_MODEL);

  // Stage the whole row in VGPRs first so all 6 b128 loads are in flight
  // before any store issues (6 loads/wave * many waves saturates HBM).
  v4f v[F4_PER_LANE];
#pragma unroll
  for (int i = 0; i < F4_PER_LANE; ++i) {
    v[i] = src[lane + 32 * i];
  }
#pragma unroll
  for (int i = 0; i < F4_PER_LANE; ++i) {
    // Non-temporal: stream to HBM, keep L2 for the embedding table.
    __builtin_nontemporal_store(v[i], &dst[lane + 32 * i]);
  }
}

extern "C" void kernel_launch(void* const* d_in, const int* in_sizes, int n_in,
                              void* d_out, int out_size, void* d_ws, size_t ws_size,
                              hipStream_t stream) {
  const int*   tokens = (const int*)d_in[0];    // [8, 4096] int32
  const float* W_E    = (const float*)d_in[1];  // [50257, 768] f32
  float*       out    = (float*)d_out;          // [8, 4096, 768] f32

  const int n_rows = in_sizes[0];               // 32768 tokens
  const int blocks = (n_rows + WAVES_PER_BLOCK - 1) / WAVES_PER_BLOCK;

  embed_gather_kernel<<<blocks, WAVES_PER_BLOCK * 32, 0, stream>>>(
      tokens, W_E, out, n_rows);
}